// AblationCam_87205015978562
// MI455X (gfx1250) — compile-verified
//
#include <hip/hip_runtime.h>

// AblationCam fused kernel for MI455X (gfx1250, wave32).
//
// Math:  out[c,p] = sum_k W[c,k]*x[k,p] + b[c]     (M=16, K=2048, N=16384)
//        logits[c] = mean_p out[c,p]
//        maps[c,p] = relu(out[c,p])                (ablation weight matrix == identity)
//
// fp32 reference -> V_WMMA_F32_16X16X4_F32 (fp32 in, fp32 acc, same precision class).
// 134 MB of x @ 23.3 TB/s ~= 5.8 us floor; 1.07 GFLOP fp32 matrix work is near the
// roofline crossover, so we run two independent WMMA accumulator chains per wave
// and 8192 waves (1024 pixel tiles x 8 K-chunks) for latency hiding.

typedef __attribute__((ext_vector_type(2))) float v2f;
typedef __attribute__((ext_vector_type(8))) float v8f;

#define CLASSES 16
#define IN_CH   2048
#define NPIX    (128 * 128)          // 16384 pixels
#define KCHUNKS 8
#define KC_LEN  (IN_CH / KCHUNKS)    // 256 channels per wave
#define WAVES_PER_BLOCK KCHUNKS      // 8 waves, one pixel tile per block
#define BLOCK_THREADS   (WAVES_PER_BLOCK * 32)  // 256 threads
#define NTILES  (NPIX / 16)          // 1024 pixel tiles == 1024 blocks

__global__ __launch_bounds__(BLOCK_THREADS)
void ablationcam_gemm(const float* __restrict__ x,     // [2048, 16384]
                      const float* __restrict__ Wm,    // [16, 2048]
                      const float* __restrict__ bias,  // [16]
                      float* __restrict__ maps,        // [16, 16384]
                      float* __restrict__ partials)    // [1024, 16]
{
    // 8 waves x (8 VGPR-rows x 32 lanes) fp32 accumulators for the K-chunk reduce
    __shared__ float red[KCHUNKS * 256];

    const int lane   = threadIdx.x & 31;
    const int wave   = threadIdx.x >> 5;   // == kchunk
    const int gtile  = blockIdx.x;         // 0..1023
    const int pbase  = gtile * 16;

    const int half = lane >> 4;            // 16-lane half
    const int lm   = lane & 15;            // A: row m (class); B: col n (pixel)
    const int k0   = half * 2;             // per-lane K offset within a K=4 step
    const int kb0  = wave * KC_LEN;

    // A 16x4 fp32: lane(m=L%16) holds K = k0, k0+1 (contiguous in W -> one b64 load)
    const float* aptr = Wm + (size_t)lm * IN_CH + (size_t)(kb0 + k0);
    // B 4x16 fp32: lane(n=L%16) holds rows k0, k0+1 -> two b32 loads,
    // each covering two 64-byte contiguous pixel segments. x streamed once -> NT.
    const float* bptr = x + (size_t)(kb0 + k0) * NPIX + (size_t)(pbase + lm);

    v8f acc0 = {};   // even K=4 steps
    v8f acc1 = {};   // odd  K=4 steps  (independent WMMA dependency chain)

    #pragma unroll 2
    for (int kb = 0; kb < KC_LEN; kb += 8) {
        v2f a0 = *(const v2f*)(aptr + kb);
        v2f a1 = *(const v2f*)(aptr + kb + 4);
        v2f b0, b1;
        b0.x = __builtin_nontemporal_load(bptr + (size_t)kb * NPIX);
        b0.y = __builtin_nontemporal_load(bptr + (size_t)(kb + 1) * NPIX);
        b1.x = __builtin_nontemporal_load(bptr + (size_t)(kb + 4) * NPIX);
        b1.y = __builtin_nontemporal_load(bptr + (size_t)(kb + 5) * NPIX);
        acc0 = __builtin_amdgcn_wmma_f32_16x16x4_f32(
                   false, a0, false, b0, (short)0, acc0, false, false);
        acc1 = __builtin_amdgcn_wmma_f32_16x16x4_f32(
                   false, a1, false, b1, (short)0, acc1, false, false);
    }

    // Spill combined accumulator to LDS for the cross-K-chunk reduction.
    #pragma unroll
    for (int r = 0; r < 8; ++r)
        red[wave * 256 + r * 32 + lane] = acc0[r] + acc1[r];
    __syncthreads();

    // Parallel epilogue: wave r owns accumulator row r
    // (C/D layout: VGPR r, lanes 0-15 -> class r, lanes 16-31 -> class r+8).
    const int c = wave + half * 8;
    float s = 0.0f;
    #pragma unroll
    for (int kc = 0; kc < KCHUNKS; ++kc)
        s += red[kc * 256 + wave * 32 + lane];
    s += bias[c];

    // coalesced write-once store: two 64B row segments per instruction
    __builtin_nontemporal_store(fmaxf(s, 0.0f),
                                &maps[(size_t)c * NPIX + (size_t)(pbase + lm)]);

    // deterministic per-tile per-class pixel sum (16-lane xor tree)
    float t = s;
    t += __shfl_xor(t, 1, 32);
    t += __shfl_xor(t, 2, 32);
    t += __shfl_xor(t, 4, 32);
    t += __shfl_xor(t, 8, 32);
    if (lm == 0)
        partials[(size_t)gtile * CLASSES + c] = t;
}

__global__ __launch_bounds__(256)
void ablationcam_finalize(const float* __restrict__ partials,  // [1024, 16]
                          float* __restrict__ logits)          // [16]
{
    __shared__ float seg[16][CLASSES];
    const int t  = threadIdx.x;
    const int c  = t & 15;
    const int sg = t >> 4;               // 16 segments of 64 tiles
    float s = 0.0f;
    for (int g = sg * 64; g < sg * 64 + 64; ++g)
        s += partials[(size_t)g * CLASSES + c];
    seg[sg][c] = s;
    __syncthreads();
    if (t < CLASSES) {
        float tot = 0.0f;
        #pragma unroll
        for (int i = 0; i < 16; ++i)
            tot += seg[i][t];
        logits[t] = tot * (1.0f / (float)NPIX);  // bias already folded in per pixel
    }
}

extern "C" void kernel_launch(void* const* d_in, const int* in_sizes, int n_in,
                              void* d_out, int out_size, void* d_ws, size_t ws_size,
                              hipStream_t stream) {
    const float* x    = (const float*)d_in[0];   // [1,2048,128,128]
    const float* Wm   = (const float*)d_in[1];   // [16,2048]
    const float* bias = (const float*)d_in[2];   // [16]

    float* logits   = (float*)d_out;             // first 16 floats
    float* maps     = (float*)d_out + CLASSES;   // next 16*16384 floats
    float* partials = (float*)d_ws;              // 1024*16 floats = 64 KB scratch

    ablationcam_gemm<<<NTILES, BLOCK_THREADS, 0, stream>>>(x, Wm, bias, maps, partials);
    ablationcam_finalize<<<1, 256, 0, stream>>>(partials, logits);
}